// HolographicField_46548855554611
// MI455X (gfx1250) — compile-verified
//
#include <hip/hip_runtime.h>
#include <math.h>

#define W_DIM 4096
#define S_DIM 4096
#define N_KEYS 1024
#define SHARPNESS 10.0f
#define CHUNK 128   // K-chunk staged in LDS for the WMMA GEMV

typedef __attribute__((ext_vector_type(2))) float v2f;
typedef __attribute__((ext_vector_type(8))) float v8f;

// ---------------------------------------------------------------------------
// Kernel 1: noisy_mix[w] = sum_s holo[w,s] * conj(cue[w,s])
//   re = hr*cr + hi*ci ; im = hi*cr - hr*ci
// One block per row w; 256 threads; float4 (b128) streaming + prefetch.
// ---------------------------------------------------------------------------
__global__ __launch_bounds__(256)
void k_noisy_mix(const float* __restrict__ hr, const float* __restrict__ hi,
                 const float* __restrict__ cr, const float* __restrict__ ci,
                 float* __restrict__ mix_re, float* __restrict__ mix_im) {
    const int w = blockIdx.x;
    const long base = (long)w * S_DIM;
    float are = 0.f, aim = 0.f;
    for (int s = threadIdx.x * 4; s < S_DIM; s += 256 * 4) {
        if (s + 1024 < S_DIM) {  // prefetch next iteration's lines
            __builtin_prefetch(hr + base + s + 1024, 0, 0);
            __builtin_prefetch(hi + base + s + 1024, 0, 0);
            __builtin_prefetch(cr + base + s + 1024, 0, 0);
            __builtin_prefetch(ci + base + s + 1024, 0, 0);
        }
        float4 h_r = *(const float4*)(hr + base + s);
        float4 h_i = *(const float4*)(hi + base + s);
        float4 c_r = *(const float4*)(cr + base + s);
        float4 c_i = *(const float4*)(ci + base + s);
        are += h_r.x * c_r.x + h_i.x * c_i.x;  aim += h_i.x * c_r.x - h_r.x * c_i.x;
        are += h_r.y * c_r.y + h_i.y * c_i.y;  aim += h_i.y * c_r.y - h_r.y * c_i.y;
        are += h_r.z * c_r.z + h_i.z * c_i.z;  aim += h_i.z * c_r.z - h_r.z * c_i.z;
        are += h_r.w * c_r.w + h_i.w * c_i.w;  aim += h_i.w * c_r.w - h_r.w * c_i.w;
    }
    // wave32 reduction
    for (int off = 16; off > 0; off >>= 1) {
        are += __shfl_down(are, off, 32);
        aim += __shfl_down(aim, off, 32);
    }
    __shared__ float sre[8], sim[8];
    const int wave = threadIdx.x >> 5, lane = threadIdx.x & 31;
    if (lane == 0) { sre[wave] = are; sim[wave] = aim; }
    __syncthreads();
    if (threadIdx.x == 0) {
        float r = 0.f, i2 = 0.f;
        for (int k = 0; k < 8; ++k) { r += sre[k]; i2 += sim[k]; }
        mix_re[w] = r; mix_im[w] = i2;
    }
}

// ---------------------------------------------------------------------------
// Kernel 2: correlations[n] = | sum_w keys[n,w] * mix[w] |  (no conjugate)
// 16 keys per block via V_WMMA_F32_16X16X4_F32.
//   acc_r = keys_re(16xK) x B,  acc_i = keys_im(16xK) x B
//   B col0 = mix_re, col1 = mix_im  (other cols 0)
//   corr_re = Dr[:,0] - Di[:,1] ; corr_im = Dr[:,1] + Di[:,0]
// Keys staged through LDS with b128 loads/stores. All 8 waves run WMMAs:
// wave v owns K-substeps [v*16, v*16+16) of each 128-wide chunk; the 8
// partial D matrices are reduced through LDS at the end.
// ---------------------------------------------------------------------------
__global__ __launch_bounds__(256)
void k_corr_wmma(const float* __restrict__ kr, const float* __restrict__ ki,
                 const float* __restrict__ mr_g, const float* __restrict__ mi_g,
                 float* __restrict__ corr) {
    __shared__ float s_kr[16 * CHUNK];
    __shared__ float s_ki[16 * CHUNK];
    __shared__ float s_mr[CHUNK];
    __shared__ float s_mi[CHUNK];
    __shared__ float s_dr[8][256];
    __shared__ float s_di[8][256];

    const int key0 = blockIdx.x * 16;
    const int tid  = threadIdx.x;
    const int wave = tid >> 5;
    const int lane = tid & 31;
    const int m    = lane & 15;          // A-matrix row == B/D column index
    const int kb   = (lane >> 4) * 2;    // K sub-offset: 0 (lanes 0-15) or 2 (16-31)

    v8f acc_r = {};
    v8f acc_i = {};

    for (int c = 0; c < W_DIM / CHUNK; ++c) {
        const int w0 = c * CHUNK;
        // Vectorized staging: 2048 floats per matrix, 256 threads * float4.
#pragma unroll
        for (int idx = tid * 4; idx < 16 * CHUNK; idx += 256 * 4) {
            const int row = idx >> 7;            // idx / CHUNK
            const int col = idx & (CHUNK - 1);   // idx % CHUNK
            const long g = (long)(key0 + row) * W_DIM + w0 + col;
            *(float4*)(s_kr + idx) = *(const float4*)(kr + g);
            *(float4*)(s_ki + idx) = *(const float4*)(ki + g);
        }
        if (tid < CHUNK / 4) {
            *(float4*)(s_mr + tid * 4) = *(const float4*)(mr_g + w0 + tid * 4);
            *(float4*)(s_mi + tid * 4) = *(const float4*)(mi_g + w0 + tid * 4);
        }
        __syncthreads();

        // Wave `wave` handles kk = wave*16 .. wave*16+15 (4 K-steps of 4).
        const int kk0 = wave * 16;
#pragma unroll
        for (int kq = 0; kq < 16; kq += 4) {
            const int kk = kk0 + kq;
            v2f a_re, a_im, b;
            a_re.x = s_kr[m * CHUNK + kk + kb];
            a_re.y = s_kr[m * CHUNK + kk + kb + 1];
            a_im.x = s_ki[m * CHUNK + kk + kb];
            a_im.y = s_ki[m * CHUNK + kk + kb + 1];
            // B[k][n]: n==0 -> mix_re[k], n==1 -> mix_im[k], else 0
            const float brx = s_mr[kk + kb],     bix = s_mi[kk + kb];
            const float bry = s_mr[kk + kb + 1], biy = s_mi[kk + kb + 1];
            b.x = (m == 0) ? brx : ((m == 1) ? bix : 0.f);
            b.y = (m == 0) ? bry : ((m == 1) ? biy : 0.f);
            acc_r = __builtin_amdgcn_wmma_f32_16x16x4_f32(
                false, a_re, false, b, (short)0, acc_r, false, false);
            acc_i = __builtin_amdgcn_wmma_f32_16x16x4_f32(
                false, a_im, false, b, (short)0, acc_i, false, false);
        }
        __syncthreads();
    }

    // Per-wave partial D matrices -> LDS.
    // D layout: VGPR r -> row r (lanes 0-15) / row r+8 (lanes 16-31), col = lane&15
    {
        const int colD   = lane & 15;
        const int rowOff = (lane >> 4) * 8;
#pragma unroll
        for (int r = 0; r < 8; ++r) {
            s_dr[wave][(rowOff + r) * 16 + colD] = acc_r[r];
            s_di[wave][(rowOff + r) * 16 + colD] = acc_i[r];
        }
    }
    __syncthreads();
    if (tid < 16) {
        float dr0 = 0.f, dr1 = 0.f, di0 = 0.f, di1 = 0.f;
#pragma unroll
        for (int v = 0; v < 8; ++v) {
            dr0 += s_dr[v][tid * 16 + 0];
            dr1 += s_dr[v][tid * 16 + 1];
            di0 += s_di[v][tid * 16 + 0];
            di1 += s_di[v][tid * 16 + 1];
        }
        const float re = dr0 - di1;
        const float im = dr1 + di0;
        corr[key0 + tid] = sqrtf(re * re + im * im);
    }
}

// ---------------------------------------------------------------------------
// Kernel 3: attention = softmax(corr * SHARPNESS), single block of 1024.
// ---------------------------------------------------------------------------
__global__ __launch_bounds__(1024)
void k_softmax(const float* __restrict__ corr, float* __restrict__ att) {
    __shared__ float red[N_KEYS];
    const int tid = threadIdx.x;
    const float x = corr[tid] * SHARPNESS;
    red[tid] = x;
    __syncthreads();
    for (int s = 512; s > 0; s >>= 1) {
        if (tid < s) red[tid] = fmaxf(red[tid], red[tid + s]);
        __syncthreads();
    }
    const float mx = red[0];
    __syncthreads();
    const float e = expf(x - mx);
    red[tid] = e;
    __syncthreads();
    for (int s = 512; s > 0; s >>= 1) {
        if (tid < s) red[tid] += red[tid + s];
        __syncthreads();
    }
    att[tid] = e / red[0];
}

// ---------------------------------------------------------------------------
// Kernel 4: clean_key[w] = sum_n attention[n] * keys[n,w]
// attention cached in LDS; column access over keys is coalesced across lanes.
// ---------------------------------------------------------------------------
__global__ __launch_bounds__(256)
void k_cleankey(const float* __restrict__ kr, const float* __restrict__ ki,
                const float* __restrict__ att,
                float* __restrict__ ckr, float* __restrict__ cki) {
    __shared__ float s_a[N_KEYS];
    const int w = blockIdx.x * 256 + threadIdx.x;
    for (int i = threadIdx.x; i < N_KEYS; i += 256) s_a[i] = att[i];
    __syncthreads();
    float r = 0.f, im = 0.f;
    for (int n = 0; n < N_KEYS; n += 4) {
#pragma unroll
        for (int j = 0; j < 4; ++j) {
            const float a = s_a[n + j];
            const long g = (long)(n + j) * W_DIM + w;
            r  += a * kr[g];
            im += a * ki[g];
        }
    }
    ckr[w] = r;
    cki[w] = im;
}

// ---------------------------------------------------------------------------
// Kernel 5: out[0] = hr*cr - hi*ci ; out[1] = hr*ci + hi*cr  (cr,ci per row)
// blockIdx: row = bx>>2, quarter-row chunk = bx&3. float4 in/out.
// ---------------------------------------------------------------------------
__global__ __launch_bounds__(256)
void k_out(const float* __restrict__ hr, const float* __restrict__ hi,
           const float* __restrict__ ckr, const float* __restrict__ cki,
           float* __restrict__ out) {
    const int row = blockIdx.x >> 2;
    const int s0  = (blockIdx.x & 3) * 1024 + threadIdx.x * 4;
    const long base = (long)row * S_DIM + s0;
    const float cr = ckr[row], ci = cki[row];
    const float4 h_r = *(const float4*)(hr + base);
    const float4 h_i = *(const float4*)(hi + base);
    float4 o_r, o_i;
    o_r.x = h_r.x * cr - h_i.x * ci;  o_i.x = h_r.x * ci + h_i.x * cr;
    o_r.y = h_r.y * cr - h_i.y * ci;  o_i.y = h_r.y * ci + h_i.y * cr;
    o_r.z = h_r.z * cr - h_i.z * ci;  o_i.z = h_r.z * ci + h_i.z * cr;
    o_r.w = h_r.w * cr - h_i.w * ci;  o_i.w = h_r.w * ci + h_i.w * cr;
    *(float4*)(out + base) = o_r;
    *(float4*)(out + (long)W_DIM * S_DIM + base) = o_i;
}

// ---------------------------------------------------------------------------
extern "C" void kernel_launch(void* const* d_in, const int* in_sizes, int n_in,
                              void* d_out, int out_size, void* d_ws, size_t ws_size,
                              hipStream_t stream) {
    const float* holo_re = (const float*)d_in[0];
    const float* holo_im = (const float*)d_in[1];
    const float* cue_re  = (const float*)d_in[2];
    const float* cue_im  = (const float*)d_in[3];
    const float* keys_re = (const float*)d_in[4];
    const float* keys_im = (const float*)d_in[5];
    float* out = (float*)d_out;

    float* ws     = (float*)d_ws;
    float* mix_re = ws;               // [4096]
    float* mix_im = ws + 4096;        // [4096]
    float* corr   = ws + 8192;        // [1024]
    float* att    = ws + 9216;        // [1024]
    float* ckr    = ws + 10240;       // [4096]
    float* cki    = ws + 14336;       // [4096]

    k_noisy_mix<<<W_DIM, 256, 0, stream>>>(holo_re, holo_im, cue_re, cue_im,
                                           mix_re, mix_im);
    k_corr_wmma<<<N_KEYS / 16, 256, 0, stream>>>(keys_re, keys_im,
                                                 mix_re, mix_im, corr);
    k_softmax<<<1, N_KEYS, 0, stream>>>(corr, att);
    k_cleankey<<<W_DIM / 256, 256, 0, stream>>>(keys_re, keys_im, att, ckr, cki);
    k_out<<<W_DIM * 4, 256, 0, stream>>>(holo_re, holo_im, ckr, cki, out);
}